// DirectRouting_26491358282215
// MI455X (gfx1250) — compile-verified
//
#include <hip/hip_runtime.h>
#include <math.h>

// Problem geometry (fixed by the reference).
#define AVALS  1152          // routing inputs A
#define BCAPS  32            // output capsules B
#define POSE   16            // 4x4 pose
#define ROWF   (BCAPS*POSE)  // 512 floats per A-row (contiguous in v)
#define ROWS   12            // A-rows per TDM chunk
#define NCHUNK (AVALS/ROWS)  // 96 chunks per pass
#define BLOCK  384           // 12 waves (wave32), wave w <-> row w of chunk

typedef float v2f  __attribute__((ext_vector_type(2)));
typedef float v8f  __attribute__((ext_vector_type(8)));
typedef unsigned int u32x4 __attribute__((ext_vector_type(4)));
typedef int   i32x4 __attribute__((ext_vector_type(4)));
typedef int   i32x8 __attribute__((ext_vector_type(8)));

// Generic pointer to an LDS object: addr[31:0] is the workgroup-relative LDS
// byte offset (LDS aperture puts the base in addr[63:32]).
__device__ __forceinline__ unsigned lds_byte_off(const void* p) {
  return (unsigned)(unsigned long long)p;
}

// ---- Tensor Data Mover: DMA a contiguous (ROWS x 512 f32) tile into LDS ----
__device__ __forceinline__ void stage_chunk(float* dst, const float* src) {
#if __has_builtin(__builtin_amdgcn_tensor_load_to_lds)
  unsigned long long ga = (unsigned long long)src;
  u32x4 g0;
  g0[0] = 1u;                                               // count=1, user D#
  g0[1] = lds_byte_off(dst);                                // lds_addr (bytes)
  g0[2] = (unsigned)ga;                                     // global_addr[31:0]
  g0[3] = (unsigned)((ga >> 32) & 0x01ffffffu) | 0x80000000u; // addr[56:32] | type=2
  i32x8 g1;
  g1[0] = (2 << 16);          // workgroup_mask=0, data_size=2 (4 bytes)
  g1[1] = (ROWF  << 16);      // tensor_dim0 = 512 (low 16 bits at [63:48])
  g1[2] = (AVALS << 16);      // tensor_dim0 hi = 0 | tensor_dim1 = 1152
  g1[3] = (ROWF  << 16);      // tensor_dim1 hi = 0 | tile_dim0 = 512
  g1[4] = ROWS;               // tile_dim1 = ROWS, tile_dim2 = 0
  g1[5] = ROWF;               // tensor_dim0_stride = 512 (contiguous rows)
  g1[6] = 0;                  // stride hi, tensor_dim1_stride lo
  g1[7] = 0;
  i32x4 g2 = {0, 0, 0, 0};    // dims 2/3 unused (tile_dim3 = 0)
  i32x4 g3 = {0, 0, 0, 0};
#if defined(__clang_major__) && (__clang_major__ >= 23)
  i32x8 g4 = {0, 0, 0, 0, 0, 0, 0, 0};
  __builtin_amdgcn_tensor_load_to_lds(g0, g1, g2, g3, g4, 0);
#else
  __builtin_amdgcn_tensor_load_to_lds(g0, g1, g2, g3, 0);
#endif
#else
  for (int i = (int)(threadIdx.x & 31); i < ROWS * ROWF; i += 32) dst[i] = src[i];
#endif
}

__device__ __forceinline__ void stage_wait() {
#if __has_builtin(__builtin_amdgcn_s_wait_tensorcnt)
  __builtin_amdgcn_s_wait_tensorcnt(0);
#endif
}

// One fused routing pass over all A. PASS=0: c=softmax(logits).
// PASS=1: c=softmax(logits+agr(p0)).  PASS=2: c=softmax(logits+agr(p0)+agr(p1)).
// Lane mapping per A-row: lane l, quarter k holds flat elements e=k*128+4l..+3,
// i.e. capsule B = 8k + (l>>2), pose p = 4*(l&3)..+3  (conflict-free b128 LDS reads).
template <int PASS>
__device__ __forceinline__ void run_pass(
    const float* __restrict__ vbatch, const float* __restrict__ logits,
    float* bufA, float* bufB, float* s_buf, float* p_buf, float* ns_lds,
    const float4 ph0[4], const float4 ph1[4])
{
  const int tid  = (int)threadIdx.x;
  const int lane = tid & 31;
  const int wave = tid >> 5;
  const int grp  = lane >> 2;

  for (int e = tid; e < ROWF; e += BLOCK) s_buf[e] = 0.0f;

  float4 acc[4];
#pragma unroll
  for (int k = 0; k < 4; ++k) acc[k] = make_float4(0.f, 0.f, 0.f, 0.f);

  // Prologue: fill both buffers (first synchronously, second stays in flight).
  if (tid < 32) {
    stage_chunk(bufA, vbatch);
    stage_wait();
    stage_chunk(bufB, vbatch + (size_t)ROWS * ROWF);
  }
  __syncthreads();

  for (int chunk = 0; chunk < NCHUNK; ++chunk) {
    const float* cur  = (chunk & 1) ? bufB : bufA;
    const int    A    = chunk * ROWS + wave;
    const float* base = cur + wave * ROWF;

    float4 vk[4];
#pragma unroll
    for (int k = 0; k < 4; ++k)
      vk[k] = *(const float4*)(base + k * 128 + lane * 4);

    float r[4];
#pragma unroll
    for (int k = 0; k < 4; ++k) {
      float d = 0.f;
      if (PASS >= 1)
        d = ph0[k].x * vk[k].x + ph0[k].y * vk[k].y +
            ph0[k].z * vk[k].z + ph0[k].w * vk[k].w;
      if (PASS == 2)
        d += ph1[k].x * vk[k].x + ph1[k].y * vk[k].y +
             ph1[k].z * vk[k].z + ph1[k].w * vk[k].w;
      if (PASS >= 1) {                 // sum the 4 pose-partials of one capsule
        d += __shfl_xor(d, 1, 32);
        d += __shfl_xor(d, 2, 32);
      }
      r[k] = logits[A * BCAPS + 8 * k + grp] + d;
    }

    // softmax over all 32 capsules (4 local + butterfly across 8 lane-groups)
    float m = fmaxf(fmaxf(r[0], r[1]), fmaxf(r[2], r[3]));
    m = fmaxf(m, __shfl_xor(m, 4, 32));
    m = fmaxf(m, __shfl_xor(m, 8, 32));
    m = fmaxf(m, __shfl_xor(m, 16, 32));
    float ex[4], se = 0.f;
#pragma unroll
    for (int k = 0; k < 4; ++k) { ex[k] = expf(r[k] - m); se += ex[k]; }
    se += __shfl_xor(se, 4, 32);
    se += __shfl_xor(se, 8, 32);
    se += __shfl_xor(se, 16, 32);
    const float inv = 1.0f / se;
#pragma unroll
    for (int k = 0; k < 4; ++k) {
      const float c = ex[k] * inv;
      acc[k].x += c * vk[k].x;  acc[k].y += c * vk[k].y;
      acc[k].z += c * vk[k].z;  acc[k].w += c * vk[k].w;
    }

    if (tid < 32) stage_wait();        // chunk+1 landed in the other buffer
    __syncthreads();                   // everyone done reading `cur`
    if (tid < 32 && chunk + 2 < NCHUNK)
      stage_chunk((chunk & 1) ? bufB : bufA,
                  vbatch + (size_t)(chunk + 2) * ROWS * ROWF);
  }

  // Reduce 12 wave-partials of s[B,p] via LDS float atomics (ds_add_f32).
#pragma unroll
  for (int k = 0; k < 4; ++k) {
    atomicAdd(&s_buf[k * 128 + lane * 4 + 0], acc[k].x);
    atomicAdd(&s_buf[k * 128 + lane * 4 + 1], acc[k].y);
    atomicAdd(&s_buf[k * 128 + lane * 4 + 2], acc[k].z);
    atomicAdd(&s_buf[k * 128 + lane * 4 + 3], acc[k].w);
  }
  __syncthreads();

  // squash: ns[p] = sum_B s[B,p]^2 == ones(16x32-chunks) x s^2 -> WMMA chain.
#if __has_builtin(__builtin_amdgcn_wmma_f32_16x16x4_f32)
  if (tid < 32) {
    const int n  = lane & 15;
    const int kh = (lane < 16) ? 0 : 2;
    v2f a; a[0] = 1.0f; a[1] = 1.0f;            // A = all-ones (layout-proof)
    v8f C = {0.f, 0.f, 0.f, 0.f, 0.f, 0.f, 0.f, 0.f};
#pragma unroll
    for (int c = 0; c < 8; ++c) {               // K covers B = 0..31
      float x0 = s_buf[(4 * c + kh + 0) * POSE + n];
      float x1 = s_buf[(4 * c + kh + 1) * POSE + n];
      v2f b; b[0] = x0 * x0; b[1] = x1 * x1;
      C = __builtin_amdgcn_wmma_f32_16x16x4_f32(false, a, false, b,
                                                (short)0, C, false, false);
    }
    if (lane < 16) ns_lds[n] = C[0];            // every row of C = ns[n]
  }
#else
  if (tid < 16) {
    float ns = 0.f;
    for (int B = 0; B < BCAPS; ++B) { float x = s_buf[B * POSE + tid]; ns += x * x; }
    ns_lds[tid] = ns;
  }
#endif
  __syncthreads();

  for (int e = tid; e < ROWF; e += BLOCK) {
    const float ns = ns_lds[e & 15];
    const float sc = sqrtf(ns + 1e-6f) / (1.0f + ns);
    p_buf[e] = sc * s_buf[e];
  }
  __syncthreads();
}

__global__ __launch_bounds__(BLOCK) void caps_routing_kernel(
    const float* __restrict__ v, const float* __restrict__ logits,
    float* __restrict__ out, int batches)
{
  __shared__ __align__(16) float bufA[ROWS * ROWF];   // 24 KB
  __shared__ __align__(16) float bufB[ROWS * ROWF];   // 24 KB
  __shared__ __align__(16) float s_buf[ROWF];
  __shared__ __align__(16) float p_buf[ROWF];
  __shared__ float ns_lds[POSE];

  const int batch = (int)blockIdx.x;
  if (batch >= batches) return;
  const float* vb = v + (size_t)batch * AVALS * ROWF;
  const int tid  = (int)threadIdx.x;
  const int lane = tid & 31;

  float4 ph0[4], ph1[4];
#pragma unroll
  for (int k = 0; k < 4; ++k) {
    ph0[k] = make_float4(0.f, 0.f, 0.f, 0.f);
    ph1[k] = make_float4(0.f, 0.f, 0.f, 0.f);
  }

  run_pass<0>(vb, logits, bufA, bufB, s_buf, p_buf, ns_lds, ph0, ph1);
#pragma unroll
  for (int k = 0; k < 4; ++k) ph0[k] = *(const float4*)&p_buf[k * 128 + lane * 4];
  run_pass<1>(vb, logits, bufA, bufB, s_buf, p_buf, ns_lds, ph0, ph1);
#pragma unroll
  for (int k = 0; k < 4; ++k) ph1[k] = *(const float4*)&p_buf[k * 128 + lane * 4];
  run_pass<2>(vb, logits, bufA, bufB, s_buf, p_buf, ns_lds, ph0, ph1);

  // Outputs: a_out (batches*32*4) first, then p (batches*32*4*4).
  const int aTotal = batches * BCAPS * 4;
  for (int e = tid; e < ROWF; e += BLOCK)
    out[aTotal + batch * ROWF + e] = p_buf[e];
  for (int t = tid; t < BCAPS * 4; t += BLOCK) {
    const int B = t >> 2, j = t & 3;
    float s2 = 0.f;
#pragma unroll
    for (int i = 0; i < 4; ++i) {        // norm over first pose axis (axis=2)
      float x = p_buf[B * POSE + i * 4 + j];
      s2 += x * x;
    }
    out[batch * (BCAPS * 4) + t] = sqrtf(s2);
  }
}

extern "C" void kernel_launch(void* const* d_in, const int* in_sizes, int n_in,
                              void* d_out, int out_size, void* d_ws, size_t ws_size,
                              hipStream_t stream) {
  (void)n_in; (void)out_size; (void)d_ws; (void)ws_size;
  const float* v      = (const float*)d_in[1];   // (batch, 1152, 32, 4, 4, 1)
  const float* logits = (const float*)d_in[2];   // (1, 1152, 32, 1, 1, 1)
  const int batches = in_sizes[0] / AVALS;       // a is (batch, 1152, 1,1,1,1)
  caps_routing_kernel<<<dim3((unsigned)batches), dim3(BLOCK), 0, stream>>>(
      v, logits, (float*)d_out, batches);
}